// Baka_84791244358183
// MI455X (gfx1250) — compile-verified
//
#include <hip/hip_runtime.h>

typedef __attribute__((ext_vector_type(2))) float v2f;
typedef __attribute__((ext_vector_type(8))) float v8f;

#define B_DIM 1024
#define I_DIM 256
#define O_DIM 512
#define W_ROW 257   // I+1 (coeff at col 0, powers at cols 1..256)
#define LDA   260   // padded LDS row stride: 260 % 64 == 4 -> conflict-free b64
#define MROWS 64    // B-rows per workgroup (4 M-tiles)

// ds_swizzle_b32 group-of-32: lane' = lane ^ m  (offset = m<<10 | and_mask 0x1f)
#define SWZ_XOR(v, m) \
    __int_as_float(__builtin_amdgcn_ds_swizzle(__float_as_int(v), ((m) << 10) | 0x1f))

__global__ __launch_bounds__(256) void powgemm_wmma_f32(
    const float* __restrict__ x,       // (B, I)
    const float* __restrict__ weight,  // (O, I, I+1)
    float* __restrict__ out)           // (B, O)
{
    __shared__ float lA[MROWS][LDA];   // log(x) tile, 64 rows of B
    __shared__ float partial[8][MROWS];

    const int o   = blockIdx.y;
    const int b0  = blockIdx.x * MROWS;
    const int tid = threadIdx.x;

    // Cooperative, coalesced fill of log(x) tile (64x256 = 16384 elems / 256 thr).
    #pragma unroll 4
    for (int it = 0; it < (MROWS * I_DIM) / 256; ++it) {
        int linear = it * 256 + tid;
        int m = linear >> 8;            // / I_DIM
        int j = linear & (I_DIM - 1);
        lA[m][j] = __logf(x[(size_t)(b0 + m) * I_DIM + j]);
    }
    __syncthreads();

    const int lane = tid & 31;
    const int wave = tid >> 5;          // 0..7: owns N-tiles {2*wave, 2*wave+1}
    const int nsel = lane & 15;         // A: M row in tile; B/C: N column in tile
    const int hi   = lane >> 4;         // 0 -> K pair {0,1}, 1 -> K pair {2,3}

    // B rows for this wave's two N-tiles: i = 32*wave + nsel and +16.
    const size_t wbase = (size_t)o * I_DIM * W_ROW;
    const float* bp = weight + wbase + (size_t)(32 * wave + nsel) * W_ROW + 1;
    const float  cf0 = bp[-1];                 // coeff[o, 32w + nsel]
    const float  cf1 = bp[16 * W_ROW - 1];     // coeff[o, 32w + 16 + nsel]

    // 8 independent accumulator chains: 4 M-tiles x 2 N-tiles.
    v8f c[4][2];
    #pragma unroll
    for (int mt = 0; mt < 4; ++mt)
        #pragma unroll
        for (int nt = 0; nt < 2; ++nt)
            c[mt][nt] = (v8f){0.f,0.f,0.f,0.f,0.f,0.f,0.f,0.f};

    #pragma unroll 2
    for (int k = 0; k < I_DIM; k += 4) {
        const int kk = k + 2 * hi;

        v2f b0v, b1v;                                   // B[kk..kk+1, n], two N-tiles
        b0v.x = bp[kk];              b0v.y = bp[kk + 1];
        b1v.x = bp[16 * W_ROW + kk]; b1v.y = bp[16 * W_ROW + kk + 1];

        v2f a[4];                                       // A[m, kk..kk+1], four M-tiles
        #pragma unroll
        for (int mt = 0; mt < 4; ++mt) {
            a[mt].x = lA[mt * 16 + nsel][kk];
            a[mt].y = lA[mt * 16 + nsel][kk + 1];
        }

        #pragma unroll
        for (int mt = 0; mt < 4; ++mt) {
            c[mt][0] = __builtin_amdgcn_wmma_f32_16x16x4_f32(
                           false, a[mt], false, b0v, (short)0, c[mt][0], false, false);
            c[mt][1] = __builtin_amdgcn_wmma_f32_16x16x4_f32(
                           false, a[mt], false, b1v, (short)0, c[mt][1], false, false);
        }
    }

    // Epilogue: fold both N-tiles before the lane reduction.
    // c[mt][nt] VGPR r holds T[b0 + mt*16 + r + 8*hi, i], i = 32w + 16*nt + nsel.
    // ds_swizzle xor-butterfly over lane bits 0..3 stays inside each 16-lane half.
    float red[4][8];
    #pragma unroll
    for (int mt = 0; mt < 4; ++mt) {
        #pragma unroll
        for (int r = 0; r < 8; ++r) {
            float s = __expf(c[mt][0][r]) * cf0 + __expf(c[mt][1][r]) * cf1;
            s += SWZ_XOR(s, 1);
            s += SWZ_XOR(s, 2);
            s += SWZ_XOR(s, 4);
            s += SWZ_XOR(s, 8);          // per-half sum over the 16 N lanes
            red[mt][r] = s;
        }
    }

    // Two exec-masked store blocks instead of 64 predicated stores.
    if (lane == 0) {
        #pragma unroll
        for (int mt = 0; mt < 4; ++mt)
            #pragma unroll
            for (int r = 0; r < 8; ++r)
                partial[wave][mt * 16 + r] = red[mt][r];
    } else if (lane == 16) {
        #pragma unroll
        for (int mt = 0; mt < 4; ++mt)
            #pragma unroll
            for (int r = 0; r < 8; ++r)
                partial[wave][mt * 16 + 8 + r] = red[mt][r];
    }
    __syncthreads();

    // Combine the 8 waves' i-range partials; one thread per output row.
    if (tid < MROWS) {
        float s = 0.f;
        #pragma unroll
        for (int w = 0; w < 8; ++w) s += partial[w][tid];
        out[(size_t)(b0 + tid) * O_DIM + o] = s;
    }
}

extern "C" void kernel_launch(void* const* d_in, const int* in_sizes, int n_in,
                              void* d_out, int out_size, void* d_ws, size_t ws_size,
                              hipStream_t stream) {
    (void)in_sizes; (void)n_in; (void)out_size; (void)d_ws; (void)ws_size;
    const float* x = (const float*)d_in[0];
    const float* w = (const float*)d_in[1];
    float* out = (float*)d_out;

    // grid.x fastest-varying -> the 16 blocks sharing an o are adjacent,
    // keeping powers[o] (256 KB) hot in L2.
    dim3 grid(B_DIM / MROWS, O_DIM);
    powgemm_wmma_f32<<<grid, 256, 0, stream>>>(x, w, out);
}